// TopKAttentionLayer_56298431316499
// MI455X (gfx1250) — compile-verified
//
#include <hip/hip_runtime.h>
#include <stdint.h>

#define SEQ    8192
#define TPB    256
#define CHUNKS 8    // 8 x float4 per thread => 32 elements/thread

typedef __attribute__((ext_vector_type(4))) float v4f;   // native vector for b128 + NT builtins

__device__ __forceinline__ uint32_t f2key(float f) {
  // order-preserving float -> uint mapping (larger float => larger key)
  uint32_t u = __float_as_uint(f);
  return (u & 0x80000000u) ? ~u : (u | 0x80000000u);
}

// lanes in my wave holding the same 8-bit digit as me (match-any via 8 ballots)
__device__ __forceinline__ uint32_t match_digit8(uint32_t digit) {
  uint32_t m = 0xFFFFFFFFu;
#pragma unroll
  for (int b = 0; b < 8; ++b) {
    uint32_t bal = __builtin_amdgcn_ballot_w32(((digit >> b) & 1u) != 0u);
    m &= ((digit >> b) & 1u) ? bal : ~bal;
  }
  return m;
}

// Descending cumulative scan of hist[256]; pick bin b* = max b with S[b] >= krem.
// Writes s_bin = b*, s_k = krem - S[b*+1]. Ends with a __syncthreads().
__device__ __forceinline__ void select_bin(int tid, int lane, int wid, uint32_t krem,
                                           uint32_t* hist, uint32_t* wtot,
                                           uint32_t* s_bin, uint32_t* s_k) {
  uint32_t v = hist[tid];
#pragma unroll
  for (int off = 1; off < 32; off <<= 1) {
    uint32_t t = __shfl_down(v, off, 32);
    if (lane + off < 32) v += t;
  }
  uint32_t wavetot = __shfl(v, 0, 32);
  if (lane == 0) wtot[wid] = wavetot;
  __syncthreads();
  uint32_t higher = 0;
  for (int w = wid + 1; w < 8; ++w) higher += wtot[w];
  uint32_t Sb  = v + higher;                            // S[tid]
  uint32_t vn  = __shfl_down(v, 1, 32);
  uint32_t Sb1 = (lane == 31) ? higher : (vn + higher); // S[tid+1], S[256]=0
  if (Sb >= krem && Sb1 < krem) {                       // unique crossing bin
    *s_bin = (uint32_t)tid & 255u;
    *s_k   = krem - Sb1;
  }
  __syncthreads();
}

__global__ __launch_bounds__(TPB) void topk_mask_kernel(
    const float* __restrict__ attn,
    const float* __restrict__ mask,
    float* __restrict__ out) {
  __shared__ __align__(16) float mask_lds[SEQ];          // 32 KB staged mask row
  __shared__ uint32_t hist[256];                         // radix histogram
  __shared__ uint32_t wtot[8];                           // per-wave bin totals
  __shared__ float    fsum[8];                           // per-wave mask sums
  __shared__ uint32_t eqm[CHUNKS][8][4];                 // equality ballots [c][wave][e]
  __shared__ uint32_t totC[CHUNKS];
  __shared__ uint32_t prefC[CHUNKS];
  __shared__ uint32_t s_bin, s_k;
  __shared__ int      s_kval;

  const int tid  = threadIdx.x;
  const int lane = tid & 31;
  const int wid  = tid >> 5;
  const size_t rowoff = (size_t)blockIdx.x * SEQ;

  // ---- async stage attn_mask row into LDS (CDNA5 async global->LDS) ----
  {
    uint32_t ldsbase = (uint32_t)(uintptr_t)(&mask_lds[0]) + (uint32_t)(tid * 16);
    uint64_t gabase  = (uint64_t)(uintptr_t)(mask + rowoff) + (uint64_t)(tid * 16);
#pragma unroll
    for (int c = 0; c < CHUNKS; ++c) {
      uint32_t lo = ldsbase + (uint32_t)(c * 4096);
      uint64_t ga = gabase + (uint64_t)(c * 4096);
      asm volatile("global_load_async_to_lds_b128 %0, %1, off"
                   :: "v"(lo), "v"(ga) : "memory");
    }
  }

  // ---- stream attn row into registers (non-temporal: no reuse), make keys ----
  uint32_t key[CHUNKS * 4];
  {
    const v4f* arow = (const v4f*)(attn + rowoff);
#pragma unroll
    for (int c = 0; c < CHUNKS; ++c) {
      v4f a = __builtin_nontemporal_load(&arow[c * 256 + tid]);
      key[c * 4 + 0] = f2key(a.x);
      key[c * 4 + 1] = f2key(a.y);
      key[c * 4 + 2] = f2key(a.z);
      key[c * 4 + 3] = f2key(a.w);
    }
  }

  // wait for this wave's async LDS fills, then sync the workgroup
  asm volatile("s_wait_asynccnt 0" ::: "memory");
  __syncthreads();

  // ---- per-row k = floor(sum(mask_row) * 0.2), clamped to [1, SEQ] ----
  {
    const v4f* mrow = (const v4f*)mask_lds;
    float s = 0.f;
#pragma unroll
    for (int c = 0; c < CHUNKS; ++c) {
      v4f m = mrow[c * 256 + tid];
      s += (m.x + m.y) + (m.z + m.w);
    }
#pragma unroll
    for (int off = 16; off > 0; off >>= 1)
      s += __shfl_down(s, off, 32);
    if (lane == 0) fsum[wid] = s;
    __syncthreads();
    if (tid == 0) {
      float t = 0.f;
      for (int w = 0; w < 8; ++w) t += fsum[w];
      int kv = (int)(t * 0.2f);
      if (kv < 1) kv = 1;
      if (kv > SEQ) kv = SEQ;
      s_kval = kv;
    }
    __syncthreads();
  }
  uint32_t krem = (uint32_t)s_kval;
  uint32_t prefix;

  // ---- radix pass 0 (top byte): heavy bin contention for normal data, so
  //      aggregate within each wave (match-any ballots, one ds_add per group) ----
  {
    hist[tid] = 0;
    __syncthreads();
#pragma unroll
    for (int j = 0; j < CHUNKS * 4; ++j) {
      uint32_t digit = key[j] >> 24;
      uint32_t m = match_digit8(digit);          // lanes sharing my digit
      if (lane == (int)__builtin_ctz(m))         // group leader
        atomicAdd(&hist[digit], (uint32_t)__popc(m));
    }
    __syncthreads();
    select_bin(tid, lane, wid, krem, hist, wtot, &s_bin, &s_k);
    prefix = s_bin & 255u;
    krem   = s_k;
    __syncthreads();
  }

  // ---- radix passes 1..3 (mantissa bytes): sparse matched subset, plain atomics ----
#pragma unroll
  for (int pass = 1; pass < 4; ++pass) {
    const int shift = 24 - 8 * pass;             // 16, 8, 0
    hist[tid] = 0;
    __syncthreads();
#pragma unroll
    for (int j = 0; j < CHUNKS * 4; ++j) {
      uint32_t kk = key[j];
      if ((kk >> (shift + 8)) == prefix)
        atomicAdd(&hist[(kk >> shift) & 255u], 1u);
    }
    __syncthreads();
    select_bin(tid, lane, wid, krem, hist, wtot, &s_bin, &s_k);
    prefix = (prefix << 8) | (s_bin & 255u);
    krem   = s_k;
    __syncthreads();
  }
  const uint32_t T = prefix;   // key of the k-th largest element
  const uint32_t r = krem;     // # of ==T elements to keep (earliest indices)

  // ---- wave32 ballots of (key == T) for stable tie ranking ----
#pragma unroll
  for (int c = 0; c < CHUNKS; ++c) {
    uint32_t b0 = __builtin_amdgcn_ballot_w32(key[c * 4 + 0] == T);
    uint32_t b1 = __builtin_amdgcn_ballot_w32(key[c * 4 + 1] == T);
    uint32_t b2 = __builtin_amdgcn_ballot_w32(key[c * 4 + 2] == T);
    uint32_t b3 = __builtin_amdgcn_ballot_w32(key[c * 4 + 3] == T);
    if (lane == 0) {
      eqm[c][wid][0] = b0; eqm[c][wid][1] = b1;
      eqm[c][wid][2] = b2; eqm[c][wid][3] = b3;
    }
  }
  __syncthreads();
  if (tid < CHUNKS) {
    uint32_t t = 0;
    for (int w = 0; w < 8; ++w)
      for (int e = 0; e < 4; ++e) t += __popc(eqm[tid][w][e]);
    totC[tid] = t;
  }
  __syncthreads();
  if (tid == 0) {
    uint32_t acc = 0;
    for (int c = 0; c < CHUNKS; ++c) { prefC[c] = acc; acc += totC[c]; }
  }
  __syncthreads();

  // ---- write mask: attn_mask at selected positions, 0 elsewhere (streaming) ----
  const uint32_t below = (lane == 0) ? 0u : (0xFFFFFFFFu >> (32 - lane));
  v4f* orow = (v4f*)(out + rowoff);
  const v4f* mrow = (const v4f*)mask_lds;
#pragma unroll
  for (int c = 0; c < CHUNKS; ++c) {
    v4f m = mrow[c * 256 + tid];
    float vals[4] = {m.x, m.y, m.z, m.w};
    float o[4];
#pragma unroll
    for (int e = 0; e < 4; ++e) {
      uint32_t kk = key[c * 4 + e];
      bool sel = kk > T;
      if (!sel && kk == T) {
        // stable rank among equal-to-threshold elements; index i = c*1024 + 4*t + e
        uint32_t rk = prefC[c];
        for (int w = 0; w < wid; ++w)
          rk += __popc(eqm[c][w][0]) + __popc(eqm[c][w][1]) +
                __popc(eqm[c][w][2]) + __popc(eqm[c][w][3]);
#pragma unroll
        for (int e2 = 0; e2 < 4; ++e2) {
          uint32_t mm = eqm[c][wid][e2];
          rk += __popc(mm & below);
          if (e2 < e) rk += (mm >> lane) & 1u;
        }
        sel = rk < r;
      }
      o[e] = sel ? vals[e] : 0.0f;
    }
    v4f ov = {o[0], o[1], o[2], o[3]};
    __builtin_nontemporal_store(ov, &orow[c * 256 + tid]);
  }
}

extern "C" void kernel_launch(void* const* d_in, const int* in_sizes, int n_in,
                              void* d_out, int out_size, void* d_ws, size_t ws_size,
                              hipStream_t stream) {
  const float* attn = (const float*)d_in[0];
  const float* mask = (const float*)d_in[1];
  float* out = (float*)d_out;
  const int bsz = in_sizes[0] / SEQ;   // 4096 rows of 8192
  topk_mask_kernel<<<dim3(bsz), dim3(TPB), 0, stream>>>(attn, mask, out);
}